// SinkhornBT_89464168775713
// MI455X (gfx1250) — compile-verified
//
#include <hip/hip_runtime.h>
#include <math.h>

// CDNA5 wave32 WMMA operand types
typedef __attribute__((ext_vector_type(2))) float v2f;   // A/B for 16x16x4 f32
typedef __attribute__((ext_vector_type(8))) float v8f;   // C/D 16x16 f32

// 4-wide log-sum-exp (max-subtracted, matches jax.nn.logsumexp numerics)
__device__ __forceinline__ float lse4(float a, float b, float c, float d) {
    float m = fmaxf(fmaxf(a, b), fmaxf(c, d));
    return m + __logf(__expf(a - m) + __expf(b - m) + __expf(c - m) + __expf(d - m));
}

// Centering transform T = 4*I - 11^T (symmetric: 3 on diag, -1 off), zero-padded to 4x16
__device__ __forceinline__ float tmat(int k, int n) {
    return (n < 4) ? ((k == n) ? 3.0f : -1.0f) : 0.0f;
}

__global__ __launch_bounds__(256) void sinkhorn_bt_kernel(
    const int* __restrict__ X, const float* __restrict__ W,
    float* __restrict__ out, int B)
{
    __shared__ float ldsW[32];       // 30-player strength table
    __shared__ float stage[256 * 4]; // strengths in -> raw_point out (per-wave reuse)

    const int t = threadIdx.x;
    int row = blockIdx.x * 256 + t;
    row = (row < B) ? row : (B - 1); // clamp, keeps EXEC all-ones for WMMA

    if (t < 30) ldsW[t] = W[t];
    __syncthreads();

    // coalesced b128 load of the 4 player indices, LDS gather of strengths
    const int4 xi = ((const int4*)X)[row];
    const float s0 = ldsW[xi.x], s1 = ldsW[xi.y], s2 = ldsW[xi.z], s3 = ldsW[xi.w];

    float raw[4];

#if __has_builtin(__builtin_amdgcn_wmma_f32_16x16x4_f32)
    {
        // ---- raw_point = S(16x4) @ T(4x16) via V_WMMA_F32_16X16X4_F32, 2 per wave ----
        stage[t * 4 + 0] = s0; stage[t * 4 + 1] = s1;
        stage[t * 4 + 2] = s2; stage[t * 4 + 3] = s3;
        __syncthreads();

        const int lane = t & 31;
        const int wave = t >> 5;
        const int n    = lane & 15; // A: row M ; B/D: column N
        const int h    = lane >> 4; // lane half: selects K pair (A/B), M+8 (D)
        const int kb   = 2 * h;     // ISA A-layout: lanes<16 -> K={0,1}, lanes>=16 -> K={2,3}

        const int rA0 = wave * 32 + n;   // wave rows 0..15
        const int rA1 = rA0 + 16;        // wave rows 16..31
        v2f a0, a1, b;
        a0.x = stage[rA0 * 4 + kb];  a0.y = stage[rA0 * 4 + kb + 1];
        a1.x = stage[rA1 * 4 + kb];  a1.y = stage[rA1 * 4 + kb + 1];
        b.x  = tmat(kb, n);          b.y  = tmat(kb + 1, n); // T symmetric -> layout-safe

        v8f c = {0.f, 0.f, 0.f, 0.f, 0.f, 0.f, 0.f, 0.f};
        v8f d0 = __builtin_amdgcn_wmma_f32_16x16x4_f32(false, a0, false, b, (short)0, c, false, false);
        v8f d1 = __builtin_amdgcn_wmma_f32_16x16x4_f32(false, a1, false, b, (short)0, c, false, false);

        __syncthreads(); // strengths consumed; reuse stage for raw_point
        if (n < 4) {     // only D columns 0..3 carry raw_point
            #pragma unroll
            for (int v = 0; v < 8; ++v) {
                const int r0 = wave * 32 + v + 8 * h;   // D layout: M = vgpr + 8*half
                stage[r0 * 4 + n]        = d0[v];
                stage[(r0 + 16) * 4 + n] = d1[v];
            }
        }
        __syncthreads();
        #pragma unroll
        for (int j = 0; j < 4; ++j) raw[j] = stage[t * 4 + j];
    }
#else
    {
        const float sum = s0 + s1 + s2 + s3;
        raw[0] = 4.f * s0 - sum; raw[1] = 4.f * s1 - sum;
        raw[2] = 4.f * s2 - sum; raw[3] = 4.f * s3 - sum;
    }
#endif

    // ---- log-domain Sinkhorn, one row per lane, all in registers ----
    // C[i][j] = (raw[i]/tau - j)^2 ; eps = 1e-3 ; loga = -log(4)
    const float kInvEps = 1.0e3f;
    const float kEps    = 1.0e-3f;
    const float kELogA  = -1.3862943611e-3f; // eps * log(1/4)

    float y[4], Cm[4][4];
    #pragma unroll
    for (int i = 0; i < 4; ++i) y[i] = raw[i] * 10.0f; // 1/tau
    #pragma unroll
    for (int i = 0; i < 4; ++i) {
        #pragma unroll
        for (int j = 0; j < 4; ++j) { float d = y[i] - (float)j; Cm[i][j] = d * d; }
    }

    float f[4] = {0.f, 0.f, 0.f, 0.f};
    float g[4] = {0.f, 0.f, 0.f, 0.f};

    for (int it = 0; it < 100; ++it) {
        #pragma unroll
        for (int i = 0; i < 4; ++i)
            f[i] = kELogA - kEps * lse4((g[0] - Cm[i][0]) * kInvEps,
                                        (g[1] - Cm[i][1]) * kInvEps,
                                        (g[2] - Cm[i][2]) * kInvEps,
                                        (g[3] - Cm[i][3]) * kInvEps);
        #pragma unroll
        for (int j = 0; j < 4; ++j)
            g[j] = kELogA - kEps * lse4((f[0] - Cm[0][j]) * kInvEps,
                                        (f[1] - Cm[1][j]) * kInvEps,
                                        (f[2] - Cm[2][j]) * kInvEps,
                                        (f[3] - Cm[3][j]) * kInvEps);
        float err = 0.f;
        #pragma unroll
        for (int i = 0; i < 4; ++i) {
            float rsum = 0.f;
            #pragma unroll
            for (int j = 0; j < 4; ++j)
                rsum += __expf((f[i] + g[j] - Cm[i][j]) * kInvEps);
            err = fmaxf(err, fabsf(rsum - 0.25f));
        }
        if (err <= 0.01f) break; // per-row stop (reference stops on batch-global max)
    }

    // expected payout: sum_j 4*P_ij * point_j ; out = raw + expected
    const float pts[4] = {-35.f, -15.f, 5.f, 45.f};
    float o[4];
    #pragma unroll
    for (int i = 0; i < 4; ++i) {
        float e = 0.f;
        #pragma unroll
        for (int j = 0; j < 4; ++j)
            e += __expf((f[i] + g[j] - Cm[i][j]) * kInvEps) * pts[j];
        o[i] = raw[i] + 4.0f * e;
    }
    ((float4*)out)[row] = make_float4(o[0], o[1], o[2], o[3]);
}

extern "C" void kernel_launch(void* const* d_in, const int* in_sizes, int n_in,
                              void* d_out, int out_size, void* d_ws, size_t ws_size,
                              hipStream_t stream) {
    (void)n_in; (void)out_size; (void)d_ws; (void)ws_size;
    const int*   X   = (const int*)d_in[0];   // [B,4] int32 player indices
    const float* W   = (const float*)d_in[1]; // [30,1] strengths
    float*       out = (float*)d_out;         // [B,4] float32
    const int B      = in_sizes[0] / 4;       // 262144 (divisible by 256)
    const int blocks = (B + 255) / 256;
    sinkhorn_bt_kernel<<<blocks, 256, 0, stream>>>(X, W, out, B);
}